// BWGNN_87943750353376
// MI455X (gfx1250) — compile-verified
//
#include <hip/hip_runtime.h>
#include <hip/hip_bf16.h>

// ---------------------------------------------------------------------------
// BWGNN forward on MI455X (gfx1250, wave32).
// Dense layers use V_WMMA_F32_16X16X4_F32 (full fp32 matrix pipe).
// Graph propagation uses f32 global atomics (HBM/atomic bound part).
// ---------------------------------------------------------------------------

typedef float v2f __attribute__((ext_vector_type(2)));
typedef float v8f __attribute__((ext_vector_type(8)));

#define N_NODES 100000
#define N_EDGES 1600000
#define HF 64

// ---------------- utility ----------------
__global__ void zero_f32(float* __restrict__ p, int n) {
    int i = blockIdx.x * blockDim.x + threadIdx.x;
    if (i < n) p[i] = 0.0f;
}

// deg[dst] += 1 for every edge
__global__ void degree_kernel(const int* __restrict__ ei, float* __restrict__ deg, int E) {
    int e = blockIdx.x * blockDim.x + threadIdx.x;
    if (e < E) atomicAdd(&deg[ei[E + e]], 1.0f);
}

// in-place: deg -> clip(deg,1)^-0.5
__global__ void dinv_kernel(float* __restrict__ d, int n) {
    int i = blockIdx.x * blockDim.x + threadIdx.x;
    if (i < n) {
        float v = fmaxf(d[i], 1.0f);
        d[i] = 1.0f / sqrtf(v);
    }
}

// ---------------- WMMA GEMM:  C[M x 64] = relu(A[M x K] @ W[K x 64] + b) ----
// blockDim = 256 (8 waves). Each wave computes one 16x16 tile.
// Block covers 32 rows x 64 cols (2 row-tiles x 4 col-tiles).
// W is staged in LDS (padded stride 65 to avoid bank conflicts).
// M must be a multiple of 32 (100000 is) so EXEC is all-1s at every WMMA.
__global__ void gemm_bias_relu_wmma(const float* __restrict__ A,
                                    const float* __restrict__ W,
                                    const float* __restrict__ bias,
                                    float* __restrict__ C,
                                    int K, int do_relu) {
    extern __shared__ float Blds[];          // K * 65 floats
    const int tid = threadIdx.x;

    for (int i = tid; i < K * HF; i += 256) {
        int kk = i >> 6, c = i & 63;
        Blds[kk * 65 + c] = W[i];
    }
    __syncthreads();

    const int wave = tid >> 5;
    const int lane = tid & 31;
    const int half = lane >> 4;              // 0: K0/K1, 1: K2/K3
    const int lm   = lane & 15;
    const int tn   = wave & 3;               // col tile 0..3
    const int tml  = wave >> 2;              // row tile within block 0..1
    const int row0 = (blockIdx.x * 2 + tml) * 16;
    const int col0 = tn * 16;

    const float* Arow = A + (long)(row0 + lm) * K;

    v8f acc = {0.f, 0.f, 0.f, 0.f, 0.f, 0.f, 0.f, 0.f};
    for (int k = 0; k < K; k += 4) {
        const int ks = k + 2 * half;
        v2f a, b;
        a.x = Arow[ks];
        a.y = Arow[ks + 1];
        b.x = Blds[ks * 65 + col0 + lm];
        b.y = Blds[(ks + 1) * 65 + col0 + lm];
        // D = A(16x4) * B(4x16) + C ; full fp32 matrix op
        acc = __builtin_amdgcn_wmma_f32_16x16x4_f32(
            false, a, false, b, (short)0, acc, false, false);
    }

    const int col  = col0 + lm;
    const float bc = bias[col];
#pragma unroll
    for (int r = 0; r < 8; ++r) {
        int row = row0 + r + 8 * half;       // C/D VGPR layout: M = r (+8 for hi half)
        float v = acc[r] + bc;
        if (do_relu) v = fmaxf(v, 0.0f);
        C[(long)row * HF + col] = v;
    }
}

// ---------------- propagation ----------------
// feat = h ; hfin[:, off:off+64] = c0 * h
__global__ void theta_init(const float* __restrict__ h, float* __restrict__ feat,
                           float* __restrict__ hfin, float c0, int off, int n64) {
    int i = blockIdx.x * blockDim.x + threadIdx.x;
    if (i >= n64) return;
    float v = h[i];
    feat[i] = v;
    int n = i >> 6, f = i & 63;
    hfin[(long)n * 192 + off + f] = c0 * v;
}

// msg[dst] += feat[src] * dinv[src]   (64 threads per edge; src/dst wave-uniform)
__global__ void edge_scatter(const int* __restrict__ ei, const float* __restrict__ feat,
                             const float* __restrict__ dinv, float* __restrict__ msg,
                             int E) {
    long long tid = (long long)blockIdx.x * blockDim.x + threadIdx.x;
    int e = (int)(tid >> 6);
    if (e >= E) return;
    int f = (int)(tid & 63);
    int s = ei[e];
    int d = ei[E + e];
    float v = feat[(long)s * HF + f] * dinv[s];
    atomicAdd(&msg[(long)d * HF + f], v);
}

// feat -= msg * dinv[node] ; hfin[:, off:off+64] += ck * feat
__global__ void theta_update(float* __restrict__ feat, const float* __restrict__ msg,
                             const float* __restrict__ dinv, float* __restrict__ hfin,
                             float ck, int off, int n64) {
    int i = blockIdx.x * blockDim.x + threadIdx.x;
    if (i >= n64) return;
    int n = i >> 6, f = i & 63;
    float v = feat[i] - msg[i] * dinv[n];
    feat[i] = v;
    hfin[(long)n * 192 + off + f] += ck * v;
}

// ---------------- final tiny layer: out[N x 2] = Hm[N x 64] @ Wm2 + bm2 -----
__global__ void final_linear(const float* __restrict__ Hm, const float* __restrict__ Wm2,
                             const float* __restrict__ bm2, float* __restrict__ out,
                             int n) {
    int i = blockIdx.x * blockDim.x + threadIdx.x;
    if (i >= n) return;
    const float* h = Hm + (long)i * HF;
    float c0 = bm2[0], c1 = bm2[1];
#pragma unroll 8
    for (int j = 0; j < HF; ++j) {
        float hv = h[j];
        c0 = fmaf(hv, Wm2[j * 2 + 0], c0);
        c1 = fmaf(hv, Wm2[j * 2 + 1], c1);
    }
    out[i * 2 + 0] = c0;
    out[i * 2 + 1] = c1;
}

// ---------------------------------------------------------------------------
extern "C" void kernel_launch(void* const* d_in, const int* in_sizes, int n_in,
                              void* d_out, int out_size, void* d_ws, size_t ws_size,
                              hipStream_t stream) {
    const float* x   = (const float*)d_in[0];
    const int*   ei  = (const int*)  d_in[1];
    const float* W1  = (const float*)d_in[2];
    const float* b1  = (const float*)d_in[3];
    const float* W2  = (const float*)d_in[4];
    const float* b2  = (const float*)d_in[5];
    const float* Wm1 = (const float*)d_in[6];
    const float* bm1 = (const float*)d_in[7];
    const float* Wm2 = (const float*)d_in[8];
    const float* bm2 = (const float*)d_in[9];
    float* out = (float*)d_out;

    const int N = N_NODES, E = N_EDGES;
    const int N64 = N * HF;

    // workspace layout (floats), with reuse:
    //   dinv : N          (degree computed in place, then inverted)
    //   h    : N*64       (post-MLP features)
    //   feat : N*64       (propagation state; reused as Hm after filters)
    //   msg  : N*64       (scatter target; reused as h1 between GEMM1/GEMM2)
    //   hfin : N*192      (concatenated filter outputs)
    float* ws   = (float*)d_ws;
    float* dinv = ws;
    float* h    = dinv + N;
    float* feat = h    + (long)N64;
    float* msg  = feat + (long)N64;
    float* hfin = msg  + (long)N64;

    // Beta-wavelet polynomial coefficients for D=2 (p/beta):
    const float TH[3][3] = {{3.0f, -3.0f, 0.75f},
                            {0.0f,  3.0f, -1.50f},
                            {0.0f,  0.0f, 0.75f}};

    // degree -> dinv
    zero_f32<<<(N + 255) / 256, 256, 0, stream>>>(dinv, N);
    degree_kernel<<<(E + 255) / 256, 256, 0, stream>>>(ei, dinv, E);
    dinv_kernel<<<(N + 255) / 256, 256, 0, stream>>>(dinv, N);

    // h1 = relu(x @ W1 + b1)   -> msg buffer
    gemm_bias_relu_wmma<<<N / 32, 256, 128 * 65 * sizeof(float), stream>>>(
        x, W1, b1, msg, 128, 1);
    // h = relu(h1 @ W2 + b2)
    gemm_bias_relu_wmma<<<N / 32, 256, 64 * 65 * sizeof(float), stream>>>(
        msg, W2, b2, h, 64, 1);

    // three Beta-wavelet filters, 2 propagation steps each
    const int eg = (int)(((long long)E * HF) / 256);   // exact: E*64 % 256 == 0
    for (int t = 0; t < 3; ++t) {
        theta_init<<<(N64 + 255) / 256, 256, 0, stream>>>(
            h, feat, hfin, TH[t][0], t * HF, N64);
        for (int k = 1; k <= 2; ++k) {
            zero_f32<<<(N64 + 255) / 256, 256, 0, stream>>>(msg, N64);
            edge_scatter<<<eg, 256, 0, stream>>>(ei, feat, dinv, msg, E);
            theta_update<<<(N64 + 255) / 256, 256, 0, stream>>>(
                feat, msg, dinv, hfin, TH[t][k], t * HF, N64);
        }
    }

    // hm = relu(hfin @ Wm1 + bm1)  -> feat buffer (reused)
    gemm_bias_relu_wmma<<<N / 32, 256, 192 * 65 * sizeof(float), stream>>>(
        hfin, Wm1, bm1, feat, 192, 1);

    // out = hm @ Wm2 + bm2
    final_linear<<<(N + 255) / 256, 256, 0, stream>>>(feat, Wm2, bm2, out, N);
}